// ExpectationLoss_80530636800416
// MI455X (gfx1250) — compile-verified
//
#include <hip/hip_runtime.h>
#include <hip/hip_bf16.h>

typedef __attribute__((ext_vector_type(2))) float v2f;
typedef __attribute__((ext_vector_type(8))) float v8f;

#define NUM_SAMPLES 64

// ---- deterministic counter-based RNG (keyed) -----------------------------
__device__ __forceinline__ unsigned hash_u32(unsigned x) {
  x ^= 42u * 0x9E3779B9u;                 // fold in the fixed key (42)
  x = x * 747796405u + 2891336453u;       // PCG-style mix
  unsigned w = ((x >> ((x >> 28) + 4u)) ^ x) * 277803737u;
  return (w >> 22) ^ w;
}

// Box-Muller pair using raw CDNA5 transcendentals:
//   v_log_f32 computes log2(x)  -> -2*ln(u1) = -2ln2 * log2(u1)
//   v_sin/cos_f32 take input in revolutions -> pass u2 directly (no 2*pi mul)
__device__ __forceinline__ void normal_pair(unsigned ctr, float& e0, float& e1) {
  const unsigned h1 = hash_u32(ctr);
  const unsigned h2 = hash_u32(ctr + 1u);
  const float u1 = ((float)h1 + 1.0f) * 2.3283064365386963e-10f;  // (0,1]
  const float u2 = (float)h2 * 2.3283064365386963e-10f;           // [0,1)
  const float r =
      __builtin_amdgcn_sqrtf(-1.3862943611198906f * __builtin_amdgcn_logf(u1));
  e0 = r * __builtin_amdgcn_cosf(u2);   // cos(2*pi*u2)
  e1 = r * __builtin_amdgcn_sinf(u2);   // sin(2*pi*u2)
}

// ---- kernel 1: per-block partial loss sums via f32 WMMA row-reduction ----
// One wave32 handles 16 rows x 64 samples. A-tile (16x4 f32, 2 VGPRs/lane):
//   lanes 0-15  hold row m=lane,    K=0,1  -> samples 4k+0, 4k+1
//   lanes 16-31 hold row m=lane-16, K=2,3  -> samples 4k+2, 4k+3
// B = ones(4x16), so D accumulates per-row sample sums in f32.
__global__ void __launch_bounds__(256)
expectation_loss_partial(const float* __restrict__ pred,
                         const float* __restrict__ td,
                         float* __restrict__ partial,
                         int B)
{
  __shared__ float s_wave[8];
  const int lane = threadIdx.x & 31;
  const int wave = threadIdx.x >> 5;
  const int wavesPerBlock = blockDim.x >> 5;      // 8
  const long long waveId = (long long)blockIdx.x * wavesPerBlock + wave;

  const int r = lane & 15;                        // row within wave's 16-row tile
  long long row = waveId * 16 + r;
  const float inRange = (row < (long long)B) ? 1.0f : 0.0f;
  if (row >= (long long)B) row = (long long)B - 1;

  const float p    = pred[row];
  const float mean = td[2 * row];
  float       sd   = td[2 * row + 1];
  const float d    = (p - mean) * inRange;        // OOB rows contribute exactly 0
  sd *= inRange;

  const int sBase = (lane >> 4) * 2;              // 0 for lanes 0-15, 2 for 16-31
  const unsigned ctrRow = (unsigned)row * (unsigned)NUM_SAMPLES;

  v8f acc = {0.f, 0.f, 0.f, 0.f, 0.f, 0.f, 0.f, 0.f};
  v2f ones; ones.x = 1.0f; ones.y = 1.0f;         // B matrix: all-ones (layout-proof)

#pragma unroll 4
  for (int k = 0; k < 16; ++k) {
    float e0, e1;
    normal_pair(ctrRow + (unsigned)(k * 4 + sBase), e0, e1);
    float l0 = d - sd * e0;
    float l1 = d - sd * e1;
    v2f a; a.x = l0 * l0; a.y = l1 * l1;
    // D = A x ones + D : accumulates 4 samples for each of 16 rows per issue
    acc = __builtin_amdgcn_wmma_f32_16x16x4_f32(
        /*neg_a=*/false, a, /*neg_b=*/false, ones,
        /*c_mod=*/(short)0, acc, /*reuse_a=*/false, /*reuse_b=*/false);
  }

  // Sum all 256 D entries = 16 * (sum of this wave's 16 row-sums),
  // independent of the exact D VGPR striping.
  float t = acc[0] + acc[1] + acc[2] + acc[3] + acc[4] + acc[5] + acc[6] + acc[7];
#pragma unroll
  for (int off = 16; off > 0; off >>= 1)
    t += __shfl_xor(t, off, 32);

  if (lane == 0) s_wave[wave] = t * (1.0f / 16.0f);
  __syncthreads();

  if (threadIdx.x == 0) {
    float bs = 0.0f;
    for (int w = 0; w < wavesPerBlock; ++w) bs += s_wave[w];
    partial[blockIdx.x] = bs;                     // deterministic tree, no atomics
  }
}

// ---- kernel 2: deterministic single-block final reduction ----------------
__global__ void __launch_bounds__(256)
reduce_partials(const float* __restrict__ partial, int n,
                float* __restrict__ out, float scale)
{
  __shared__ float sh[256];
  float t = 0.0f;
  for (int i = threadIdx.x; i < n; i += 256) t += partial[i];
  sh[threadIdx.x] = t;
  __syncthreads();
#pragma unroll
  for (int s = 128; s > 0; s >>= 1) {
    if (threadIdx.x < (unsigned)s) sh[threadIdx.x] += sh[threadIdx.x + s];
    __syncthreads();
  }
  if (threadIdx.x == 0) out[0] = sh[0] * scale;
}

extern "C" void kernel_launch(void* const* d_in, const int* in_sizes, int n_in,
                              void* d_out, int out_size, void* d_ws, size_t ws_size,
                              hipStream_t stream) {
  const float* pred = (const float*)d_in[0];      // (B,1) f32
  const float* td   = (const float*)d_in[1];      // (B,2) f32: [mean, std]
  const int B = in_sizes[0];

  float* partial = (float*)d_ws;                  // one f32 per block (64 KB @ B=2M)

  const int rowsPerBlock = 128;                   // 8 waves x 16 rows
  const int blocks = (B + rowsPerBlock - 1) / rowsPerBlock;

  expectation_loss_partial<<<blocks, 256, 0, stream>>>(pred, td, partial, B);

  const float scale = (float)(1.0 / ((double)B * (double)NUM_SAMPLES));
  reduce_partials<<<1, 256, 0, stream>>>(partial, blocks, (float*)d_out, scale);
}